// Treeish_75007308857860
// MI455X (gfx1250) — compile-verified
//
#include <hip/hip_runtime.h>
#include <hip/hip_bf16.h>

typedef __attribute__((ext_vector_type(2))) float v2f;
typedef __attribute__((ext_vector_type(8))) float v8f;

#define TS_LATENT 4096
#define TS_PART_STRIDE 80   // 64 t-cols + u at [64]; padded to 80
#define TS_PART_OFF 128     // partials start 128 floats into ws (d/c0 live below)

// ---------------------------------------------------------------------------
// Prep: ws[s] = d[s] = sum_f bk[f]*W[s,f]  (s = 0..63),  ws[64] = c0 = sum bq
// ---------------------------------------------------------------------------
__global__ __launch_bounds__(128) void ts_prep(const float* __restrict__ bq,
                                               const float* __restrict__ bk,
                                               const float* __restrict__ W,
                                               float* __restrict__ ws, int F) {
  const int s = threadIdx.x;
  if (s < 64) {
    const float* wr = W + (size_t)s * F;
    float acc = 0.f;
    for (int f = 0; f < F; ++f) acc = fmaf(bk[f], wr[f], acc);
    ws[s] = acc;
  } else if (s == 64) {
    float acc = 0.f;
    for (int f = 0; f < F; ++f) acc += bq[f];
    ws[64] = acc;
  }
}

static __device__ __forceinline__ v8f wmma_f32(v2f a, v2f b, v8f c) {
  // D = A(16x4 f32) * B(4x16 f32) + C(16x16 f32)
  return __builtin_amdgcn_wmma_f32_16x16x4_f32(false, a, false, b, (short)0, c,
                                               false, false);
}

// ---------------------------------------------------------------------------
// Main: one wave = one 16-row batch tile x S=64 (4 WMMA N-tiles), over the
// K-range [kbeg,kend) selected by blockIdx.y (split-K). FUSED=true applies the
// nonlinear epilogue directly; FUSED=false writes deterministic partials.
// ---------------------------------------------------------------------------
template <bool FUSED>
__global__ __launch_bounds__(128) void ts_main(
    const float* __restrict__ z, const float* __restrict__ wq,
    const float* __restrict__ wk, const float* __restrict__ W,
    const float* __restrict__ bfin, const int* __restrict__ idx,
    const float* __restrict__ ws, float* __restrict__ part,
    float* __restrict__ out, int F, int batch, int kchunk) {
  const int lane  = threadIdx.x & 31;
  const int wave  = threadIdx.x >> 5;
  const int m0    = (blockIdx.x * 4 + wave) * 16;
  if (m0 >= batch) return;                 // uniform per wave: EXEC stays full
  const int split = blockIdx.y;
  const int kbeg  = split * kchunk;
  const int kend  = (kbeg + kchunk < F) ? (kbeg + kchunk) : F;
  if (kbeg >= kend) return;

  const int col   = lane & 15;             // N within tile / A-row within tile
  const int khalf = lane >> 4;             // K-half selector (A/B layout)

  int mrow = m0 + col;
  if (mrow >= batch) mrow = batch - 1;     // clamp reads; writes guarded below
  const float* zrow = z + (size_t)mrow * TS_LATENT;

  const float* Wr0 = W + (size_t)(col +  0) * F;
  const float* Wr1 = W + (size_t)(col + 16) * F;
  const float* Wr2 = W + (size_t)(col + 32) * F;
  const float* Wr3 = W + (size_t)(col + 48) * F;

  v8f acc0 = {}, acc1 = {}, acc2 = {}, acc3 = {};
  float uacc = 0.f;

  const int kmain = kbeg + ((kend - kbeg) & ~3);
#pragma unroll 4
  for (int k = kbeg; k < kmain; k += 4) {
    const int kk = k + 2 * khalf;
    const int2 ip = *(const int2*)(idx + kk);      // 8B-aligned (kk even)
    v2f a;
    a.x = zrow[ip.x];                              // gathered A fragment
    a.y = zrow[ip.y];
    const v2f wk2 = *(const v2f*)(wk + kk);
    const v2f wq2 = *(const v2f*)(wq + kk);
    v2f b0, b1, b2, b3;                            // B = wk ⊙ W on the fly
    b0.x = Wr0[kk] * wk2.x;  b0.y = Wr0[kk + 1] * wk2.y;
    b1.x = Wr1[kk] * wk2.x;  b1.y = Wr1[kk + 1] * wk2.y;
    b2.x = Wr2[kk] * wk2.x;  b2.y = Wr2[kk + 1] * wk2.y;
    b3.x = Wr3[kk] * wk2.x;  b3.y = Wr3[kk + 1] * wk2.y;
    uacc = fmaf(a.x, wq2.x, fmaf(a.y, wq2.y, uacc));  // u[b] on VALU
    acc0 = wmma_f32(a, b0, acc0);
    acc1 = wmma_f32(a, b1, acc1);
    acc2 = wmma_f32(a, b2, acc2);
    acc3 = wmma_f32(a, b3, acc3);
  }

  // K-range % 4 tail: clamp indices, zero weight scalars => exact 0 contrib
  if (kmain < kend) {
    const int kk = kmain + 2 * khalf;
    const bool v0 = kk < kend;
    const bool v1 = (kk + 1) < kend;
    const int kc0 = v0 ? kk : kbeg;
    const int kc1 = v1 ? (kk + 1) : kbeg;
    v2f a;
    a.x = zrow[idx[kc0]];
    a.y = zrow[idx[kc1]];
    const float wkx = v0 ? wk[kc0] : 0.f;
    const float wky = v1 ? wk[kc1] : 0.f;
    const float wqx = v0 ? wq[kc0] : 0.f;
    const float wqy = v1 ? wq[kc1] : 0.f;
    v2f b0, b1, b2, b3;
    b0.x = Wr0[kc0] * wkx;  b0.y = Wr0[kc1] * wky;
    b1.x = Wr1[kc0] * wkx;  b1.y = Wr1[kc1] * wky;
    b2.x = Wr2[kc0] * wkx;  b2.y = Wr2[kc1] * wky;
    b3.x = Wr3[kc0] * wkx;  b3.y = Wr3[kc1] * wky;
    uacc = fmaf(a.x, wqx, fmaf(a.y, wqy, uacc));
    acc0 = wmma_f32(a, b0, acc0);
    acc1 = wmma_f32(a, b1, acc1);
    acc2 = wmma_f32(a, b2, acc2);
    acc3 = wmma_f32(a, b3, acc3);
  }

  // Row-sum u: lane l and l^16 hold the two K-half partials of row (l&15)
  const float usum = uacc + __shfl_xor(uacc, 16, 32);

  if constexpr (FUSED) {
    const float c0 = ws[64];
    const float d0 = ws[col +  0], d1 = ws[col + 16];
    const float d2 = ws[col + 32], d3 = ws[col + 48];
    const float f0 = bfin[col +  0], f1 = bfin[col + 16];
    const float f2 = bfin[col + 32], f3 = bfin[col + 48];
#pragma unroll
    for (int i = 0; i < 8; ++i) {
      const int rloc = i + 8 * khalf;            // C/D layout: M = i + 8*(l>=16)
      const float u  = __shfl(usum, rloc, 32);   // broadcast row's u
      const float sb = u + c0;
      const int row  = m0 + rloc;
      if (row < batch) {
        float* orow = out + (size_t)row * 64;
        orow[col +  0] = fmaf(sb, acc0[i] + d0, f0);
        orow[col + 16] = fmaf(sb, acc1[i] + d1, f1);
        orow[col + 32] = fmaf(sb, acc2[i] + d2, f2);
        orow[col + 48] = fmaf(sb, acc3[i] + d3, f3);
      }
    }
  } else {
    float* pbase = part + (size_t)split * batch * TS_PART_STRIDE;
#pragma unroll
    for (int i = 0; i < 8; ++i) {
      const int rloc = i + 8 * khalf;
      const int row  = m0 + rloc;
      if (row < batch) {
        float* prow = pbase + (size_t)row * TS_PART_STRIDE;
        prow[col +  0] = acc0[i];
        prow[col + 16] = acc1[i];
        prow[col + 32] = acc2[i];
        prow[col + 48] = acc3[i];
      }
    }
    if (lane < 16 && (m0 + lane) < batch)        // usum in lane l is row l&15
      pbase[(size_t)(m0 + lane) * TS_PART_STRIDE + 64] = usum;
  }
}

// ---------------------------------------------------------------------------
// Split-K combine + nonlinear epilogue (fixed-order sum => deterministic)
// ---------------------------------------------------------------------------
__global__ __launch_bounds__(256) void ts_epi(const float* __restrict__ part,
                                              const float* __restrict__ ws,
                                              const float* __restrict__ bfin,
                                              float* __restrict__ out,
                                              int batch, int KS) {
  const int gid = blockIdx.x * 256 + threadIdx.x;
  const int row = gid >> 6;
  const int col = gid & 63;
  if (row >= batch) return;
  float t = 0.f, u = 0.f;
  for (int s = 0; s < KS; ++s) {
    const float* p = part + ((size_t)s * batch + row) * TS_PART_STRIDE;
    t += p[col];
    u += p[64];
  }
  out[(size_t)row * 64 + col] = fmaf(u + ws[64], t + ws[col], bfin[col]);
}

// ---------------------------------------------------------------------------
extern "C" void kernel_launch(void* const* d_in, const int* in_sizes, int n_in,
                              void* d_out, int out_size, void* d_ws,
                              size_t ws_size, hipStream_t stream) {
  const float* z  = (const float*)d_in[0];
  const float* wq = (const float*)d_in[1];
  const float* bq = (const float*)d_in[2];
  const float* wk = (const float*)d_in[3];
  const float* bk = (const float*)d_in[4];
  const float* W  = (const float*)d_in[5];
  const float* bf = (const float*)d_in[6];
  const int*   idx = (const int*)d_in[7];
  const int F     = in_sizes[1];              // masked feature count
  const int batch = in_sizes[0] / TS_LATENT;  // 16384
  float* ws   = (float*)d_ws;                 // [0..64]: d, c0
  float* part = ws + TS_PART_OFF;
  float* out  = (float*)d_out;

  // Pick largest split-K whose partial buffer fits in ws (pure fn of ws_size).
  auto need = [&](int s) {
    return (size_t)TS_PART_OFF * 4 + (size_t)s * batch * TS_PART_STRIDE * 4;
  };
  int KS = 1;
  if (ws_size >= need(4)) KS = 4;
  else if (ws_size >= need(2)) KS = 2;

  ts_prep<<<1, 128, 0, stream>>>(bq, bk, W, ws, F);

  const int mtiles = (batch + 15) / 16;
  const int blocks = (mtiles + 3) / 4;        // 4 waves (128 thr) per block
  const int kchunk = (((F + KS - 1) / KS) + 3) & ~3;   // multiple of 4

  if (KS == 1) {
    dim3 grid(blocks, 1);
    ts_main<true><<<grid, 128, 0, stream>>>(z, wq, wk, W, bf, idx, ws, part,
                                            out, F, batch, kchunk);
  } else {
    dim3 grid(blocks, KS);
    ts_main<false><<<grid, 128, 0, stream>>>(z, wq, wk, W, bf, idx, ws, part,
                                             out, F, batch, kchunk);
    const int n = batch * 64;
    ts_epi<<<(n + 255) / 256, 256, 0, stream>>>(part, ws, bf, out, batch, KS);
  }
}